// MultiEmbedding_25589415149888
// MI455X (gfx1250) — compile-verified
//
#include <hip/hip_runtime.h>
#include <hip/hip_bf16.h>
#include <stdint.h>

// out[n, :] = sum_{l=0..7} w[l, x[n,l], :]
// N=65536, L=8, K=1024, D=1024, fp32.
//
// Memory-bound kernel (0.47 GFLOP vs 256 MiB output stream + 2 GiB gather):
//  - w (32 MiB) is L2-resident on MI455X (192 MB L2), ~64x row reuse.
//  - output (256 MiB) is write-once -> non-temporal stores so it never
//    evicts the table from L2. HBM floor: 256 MiB / 23.3 TB/s ~= 11 us.
//  - WMMA intentionally unused: the one-hot matmul formulation would be
//    ~550 TFLOP of mostly-zero MACs, ~1000x the memory-bound cost.
// Data movement uses the gfx1250 async global->LDS path (ASYNCcnt).

typedef float v4f __attribute__((ext_vector_type(4)));

__global__ __launch_bounds__(256)
void MultiEmbedding_25589415149888_kernel(const int* __restrict__ x,
                                          const float* __restrict__ w,
                                          float* __restrict__ out) {
    constexpr int L = 8;
    constexpr int K = 1024;
    constexpr int D = 1024;

    const unsigned n = blockIdx.x;     // output row
    const unsigned t = threadIdx.x;    // 0..255, each lane owns 4 floats of the row

    __shared__ __align__(16) float buf[L * D];   // 8 rows x 4 KiB = 32 KiB LDS

    // Fetch all 8 block-uniform indices with two wide scalar loads and ONE
    // kmcnt wait, instead of 8 serialized s_load_b32 / s_wait_kmcnt pairs.
    const int4* __restrict__ xr = reinterpret_cast<const int4*>(x + (size_t)n * L);
    const int4 i03 = xr[0];
    const int4 i47 = xr[1];
    const int idx[L] = { i03.x, i03.y, i03.z, i03.w,
                         i47.x, i47.y, i47.z, i47.w };

    const unsigned lane_byte = t * 16u;               // 16 B per lane

    // Issue 8 async 128b copies per lane back-to-back: global (L2-resident)
    // -> LDS. Tracked by ASYNCcnt; no VGPR staging, no barrier needed since
    // each lane consumes exactly the LDS bytes it requested.
#pragma unroll
    for (int l = 0; l < L; ++l) {
        const unsigned goff = ((unsigned)(l * K) + (unsigned)idx[l]) * (D * 4u)
                              + lane_byte;                                      // byte offset into w
        const unsigned loff = (unsigned)(uintptr_t)(&buf[l * D]) + lane_byte;   // LDS byte offset
        asm volatile("global_load_async_to_lds_b128 %0, %1, %2"
                     :
                     : "v"(loff), "v"(goff), "s"(w)
                     : "memory");
    }

    // Wait once for all async copies from this wave.
    asm volatile("s_wait_asynccnt 0" ::: "memory");

    // Accumulate the 8 gathered rows (8 x ds_load_b128 + packed VALU adds).
    v4f acc = (v4f)0.0f;
#pragma unroll
    for (int l = 0; l < L; ++l) {
        const v4f v = *reinterpret_cast<const v4f*>(&buf[l * D + 4 * t]);
        acc += v;
    }

    // Non-temporal 128b store: keep the 256 MiB output stream out of L2 so
    // the 32 MiB embedding table stays resident.
    v4f* o = reinterpret_cast<v4f*>(out + (size_t)n * D) + t;
    __builtin_nontemporal_store(acc, o);
}

extern "C" void kernel_launch(void* const* d_in, const int* in_sizes, int n_in,
                              void* d_out, int out_size, void* d_ws, size_t ws_size,
                              hipStream_t stream) {
    const int*   x   = (const int*)d_in[0];   // (N, 8) indices
    const float* w   = (const float*)d_in[1]; // (8, 1024, 1024) fp32
    float*       out = (float*)d_out;         // (N, 1024) fp32

    const int N = in_sizes[0] / 8;            // 65536

    dim3 grid((unsigned)N);
    dim3 block(256);
    MultiEmbedding_25589415149888_kernel<<<grid, block, 0, stream>>>(x, w, out);
}